// GPTConvTranspose2d_3736621547828
// MI455X (gfx1250) — compile-verified
//
#include <hip/hip_runtime.h>

typedef __attribute__((ext_vector_type(2))) float v2f;
typedef __attribute__((ext_vector_type(8))) float v8f;

#define IC 256
#define OC 128
#define IH 32
#define IW 32
#define OH 66
#define OW 66
#define JMAX 33          // per-parity sub-grid is 33x33
#define XSROW 17         // staged cols per (ic, row): x in [i0-1, i0+15]

// One workgroup = 8 waves (256 thr). All waves share TWO 16-pixel tiles
// (rows j and j+1 of one parity class); wave w computes oc block [16w,16w+16)
// via WMMA f32 16x16x4. K = the 4 taps (dy,dx) of one input channel; each
// weight fragment (A) is reused by both row tiles -> 2 WMMAs per A load.
// K order: k=2*half + {0: dx=1, 1: dx=0}, dy = half  (matches ds_load_2addr
// return order so B fragments need no register swaps).
// LDS layout [ic][row][xo]: row/ic deltas (17, 51 elems) fit ds immediate
// offsets -> all 8 ds loads per unrolled body share one base register.
__global__ __launch_bounds__(256)
void convT2d_wmma_f32(const float* __restrict__ x,
                      const float* __restrict__ w,
                      const float* __restrict__ bias,
                      float* __restrict__ out)
{
    __shared__ float xs[IC * 3 * XSROW];   // [ic][row: iy=j-1+row][xo]

    int bid = blockIdx.x;
    int it  = bid % 3;   bid /= 3;         // i-tile: i0 = 0,16,32
    int jt  = bid % 17;  bid /= 17;        // row pair: j = 2*jt, j+1
    int b   = bid % 16;  bid /= 16;        // batch
    int py  = bid & 1;                     // output parity y
    int px  = bid >> 1;                    // output parity x
    int j   = 2 * jt;
    int i0  = it * 16;
    int tid = threadIdx.x;

    // ---- Stage activation rows iy = j-1, j, j+1 for cols x in [i0-1, i0+15];
    // ---- out-of-bounds baked to zero so the hot loop needs no masks.
    for (int e = tid; e < IC * 3 * XSROW; e += 256) {
        int ic  = e / (3 * XSROW);
        int r   = e - ic * (3 * XSROW);
        int row = r / XSROW;
        int xo  = r - row * XSROW;
        int iy  = j - 1 + row;
        int xx  = i0 - 1 + xo;
        float v = 0.0f;
        if (iy >= 0 && iy < IH && xx >= 0 && xx < IW)
            v = x[((b * IC + ic) * IH + iy) * IW + xx];
        xs[e] = v;
    }
    __syncthreads();

    int wv   = tid >> 5;        // wave id -> oc block
    int lane = tid & 31;
    int half = lane >> 4;       // K-half: dy = half
    int nn   = lane & 15;       // A: M row; B: N col

    // A-fragment: lane holds K = 2*half + {0,1} -> (kx=px+2, kx=px), ky=py+2*half.
    // px baked into the pointer: a.x = wq[2] (dx=1), a.y = wq[0] (dx=0).
    int ky = py + 2 * half;
    const float* wq = w + ((wv * 16 + nn) * 16 + ky * 4 + px);   // +2048 per ic

    // B-fragment: tile0 row = 1-half (iy = j-half); tile1 = tile0 + XSROW.
    // b.x = xs[..][nn]   (x = i-1, dx=1)  = K even
    // b.y = xs[..][nn+1] (x = i,   dx=0)  = K odd
    const float* lp = &xs[(1 - half) * XSROW + nn];              // +3*XSROW per ic

    v8f accA0 = {0.f,0.f,0.f,0.f,0.f,0.f,0.f,0.f};   // tile0, even ic
    v8f accA1 = accA0;                               // tile0, odd ic
    v8f accB0 = accA0;                               // tile1, even ic
    v8f accB1 = accA0;                               // tile1, odd ic

    #pragma unroll 2
    for (int ic = 0; ic < IC; ic += 2) {
        v2f a0, a1, b00, b01, b10, b11;
        a0.x = wq[2];        a0.y = wq[0];            // ic
        a1.x = wq[2048 + 2]; a1.y = wq[2048];         // ic+1
        b00.x = lp[0];                 b00.y = lp[1];                  // tile0, ic
        b01.x = lp[XSROW];             b01.y = lp[XSROW + 1];          // tile1, ic
        b10.x = lp[3 * XSROW];         b10.y = lp[3 * XSROW + 1];      // tile0, ic+1
        b11.x = lp[4 * XSROW];         b11.y = lp[4 * XSROW + 1];      // tile1, ic+1
        wq += 2 * 2048;
        lp += 6 * XSROW;

        accA0 = __builtin_amdgcn_wmma_f32_16x16x4_f32(
                    false, a0, false, b00, (short)0, accA0, false, false);
        accB0 = __builtin_amdgcn_wmma_f32_16x16x4_f32(
                    false, a0, false, b01, (short)0, accB0, false, false);
        accA1 = __builtin_amdgcn_wmma_f32_16x16x4_f32(
                    false, a1, false, b10, (short)0, accA1, false, false);
        accB1 = __builtin_amdgcn_wmma_f32_16x16x4_f32(
                    false, a1, false, b11, (short)0, accB1, false, false);
    }

    // ---- Epilogue. D layout: VGPR r, lanes0-15 -> M=r, lanes16-31 -> M=8+r.
    // One base pointer; per-r oc stride (17424 B) and tile1 row stride (528 B)
    // fold into 24-bit instruction offsets.
    int ii = i0 + nn;
    if (ii < JMAX) {
        int oc0 = wv * 16 + half * 8;
        const float* bp = bias + oc0;
        float* p = out + ((size_t)(b * OC + oc0) * OH + (2 * j + py)) * OW
                       + (2 * ii + px);
        bool t1ok = (j + 1) < JMAX;
        #pragma unroll
        for (int r = 0; r < 8; ++r) {
            float bs = bp[r];
            p[r * (OH * OW)] = accA0[r] + accA1[r] + bs;
            if (t1ok)
                p[r * (OH * OW) + 2 * OW] = accB0[r] + accB1[r] + bs;
        }
    }
}

extern "C" void kernel_launch(void* const* d_in, const int* in_sizes, int n_in,
                              void* d_out, int out_size, void* d_ws, size_t ws_size,
                              hipStream_t stream) {
    const float* x    = (const float*)d_in[0];   // (16,256,32,32)
    const float* wgt  = (const float*)d_in[1];   // (256,128,4,4)
    const float* bias = (const float*)d_in[2];   // (128,)
    float* out = (float*)d_out;                  // (16,128,66,66)

    // 4 parity classes * 16 batches * 17 row-pairs * 3 i-tiles
    dim3 grid(4 * 16 * 17 * 3);
    convT2d_wmma_f32<<<grid, dim3(256), 0, stream>>>(x, wgt, bias, out);
}